// GCNLinkPrediction_2869038154062
// MI455X (gfx1250) — compile-verified
//
#include <hip/hip_runtime.h>
#include <math.h>

typedef float v2f __attribute__((ext_vector_type(2)));
typedef float v8f __attribute__((ext_vector_type(8)));

#define IN_F  128
#define HID_F 100
#define OUT_F 64
#define HID_PAD 112   // 7 * 16

// ---------------- degree / normalization ----------------

__global__ void k_deg_init(int* __restrict__ deg, int n) {
    int i = blockIdx.x * blockDim.x + threadIdx.x;
    if (i < n) deg[i] = 1;                       // self-loop contributes 1
}

__global__ void k_deg_scatter(const int* __restrict__ dst, int E, int* __restrict__ deg) {
    int e = blockIdx.x * blockDim.x + threadIdx.x;
    if (e < E) atomicAdd(&deg[dst[e]], 1);
}

__global__ void k_dinv(const int* __restrict__ deg, float* __restrict__ dinv, int n) {
    int i = blockIdx.x * blockDim.x + threadIdx.x;
    if (i < n) dinv[i] = rsqrtf((float)deg[i]);  // deg >= 1 always (self-loop)
}

// ---------------- weight prep: transpose (+ zero-pad cols) ----------------
// Bt[c][k] = (c < ncols) ? B[k][c] : 0   — makes per-lane B fragments contiguous
// (one b64 load) and removes every runtime bound from the GEMM inner loop.
__global__ void k_transpose_pad(const float* __restrict__ B, float* __restrict__ Bt,
                                int K, int ncols, int ncols_pad) {
    int idx = blockIdx.x * blockDim.x + threadIdx.x;
    int total = ncols_pad * K;
    if (idx >= total) return;
    int c = idx / K;
    int k = idx - c * K;
    Bt[idx] = (c < ncols) ? B[(size_t)k * ncols + c] : 0.0f;
}

// ---------------- WMMA fp32 GEMM: C[M x NCOLS] = A[M x K] * B[K x NCOLS] -----
// B supplied transposed+padded as Bt[NT*16 x K]. One wave per 16-row tile; NT
// column tiles of 16 share each A fragment. K, NCOLS compile-time: no guards in
// the hot loop; straight-line global_load_b64 + v_wmma_f32_16x16x4_f32.
template <int NT, int K, int NCOLS>
__global__ __launch_bounds__(32)
void k_gemm_wmma(const float* __restrict__ A, const float* __restrict__ Bt,
                 float* __restrict__ C, int nrows) {
    const int lane = threadIdx.x & 31;
    const int half = lane >> 4;        // 0: lanes 0-15 (K=0,1), 1: lanes 16-31 (K=2,3)
    const int l15  = lane & 15;
    const int rowBase = blockIdx.x * 16;

    v8f acc[NT] = {};

    const int m = min(rowBase + l15, nrows - 1);
    const float* ap = A + (size_t)m * K + 2 * half;        // 8B-aligned (K even)
    const float* bp[NT];
#pragma unroll
    for (int t = 0; t < NT; ++t)
        bp[t] = Bt + (size_t)(t * 16 + l15) * K + 2 * half;

#pragma unroll 4
    for (int kk = 0; kk < K; kk += 4) {
        const v2f a = *(const v2f*)(ap + kk);              // A[m][kk+2h .. kk+2h+1]
#pragma unroll
        for (int t = 0; t < NT; ++t) {
            const v2f b = *(const v2f*)(bp[t] + kk);       // B[kk+2h..][col] contiguous in Bt
            acc[t] = __builtin_amdgcn_wmma_f32_16x16x4_f32(
                false, a, false, b, (short)0, acc[t], false, false);
        }
    }

    // C 16x16 f32 layout: lanes 0-15 -> N=l15, rows rowBase+i; lanes 16-31 -> rows rowBase+8+i
#pragma unroll
    for (int t = 0; t < NT; ++t) {
        const int col = t * 16 + l15;
        if (NCOLS == NT * 16 || col < NCOLS) {             // compile-time for layer 2
#pragma unroll
            for (int i = 0; i < 8; ++i) {
                const int r = rowBase + half * 8 + i;
                if (r < nrows) C[(size_t)r * NCOLS + col] = acc[t][i];
            }
        }
    }
}

// ---------------- aggregation ----------------

// agg[i][c] = h[i][c] * dinv[i]^2 (self-loop term; full overwrite => no memset)
__global__ void k_selfloop(const float* __restrict__ h, const float* __restrict__ dinv,
                           float* __restrict__ agg, int n, int C) {
    long long total = (long long)n * C;
    for (long long idx = (long long)blockIdx.x * blockDim.x + threadIdx.x; idx < total;
         idx += (long long)gridDim.x * blockDim.x) {
        int i = (int)(idx / C);
        float s = dinv[i];
        agg[idx] = h[idx] * s * s;
    }
}

// one edge per block; channel-per-lane -> coalesced gather + coalesced L2 atomics
__global__ void k_edge_scatter(const int* __restrict__ src, const int* __restrict__ dst,
                               const float* __restrict__ h, const float* __restrict__ dinv,
                               float* __restrict__ agg, int C) {
    const int e = blockIdx.x;
    const int c = threadIdx.x;
    const int s = src[e];              // uniform per block -> scalar loads
    const int d = dst[e];
    const float w = dinv[s] * dinv[d];
    if (c < C) {
        atomicAdd(&agg[(size_t)d * C + c], h[(size_t)s * C + c] * w);
    }
}

__global__ void k_bias_relu(float* __restrict__ agg, const float* __restrict__ bias,
                            int n, int C) {
    long long total = (long long)n * C;
    for (long long idx = (long long)blockIdx.x * blockDim.x + threadIdx.x; idx < total;
         idx += (long long)gridDim.x * blockDim.x) {
        float v = agg[idx] + bias[idx % C];
        agg[idx] = v > 0.0f ? v : 0.0f;
    }
}

// out[i][:] = (agg[i][:] + b2) / max(||.||_2, 1e-12); one wave per node,
// 2 channels per lane (64 = 2*warpSize), warp shuffle reduction.
__global__ void k_bias_norm(const float* __restrict__ agg, const float* __restrict__ bias,
                            float* __restrict__ out, int n) {
    const int wid  = threadIdx.x >> 5;
    const int lane = threadIdx.x & 31;
    const int node = blockIdx.x * (blockDim.x >> 5) + wid;
    if (node >= n) return;
    const size_t base = (size_t)node * OUT_F;
    float v0 = agg[base + lane]      + bias[lane];
    float v1 = agg[base + 32 + lane] + bias[32 + lane];
    float s = v0 * v0 + v1 * v1;
#pragma unroll
    for (int off = 16; off > 0; off >>= 1) s += __shfl_xor(s, off, 32);
    const float inv = 1.0f / fmaxf(sqrtf(s), 1e-12f);
    out[base + lane]      = v0 * inv;
    out[base + 32 + lane] = v1 * inv;
}

// ---------------- launch ----------------

extern "C" void kernel_launch(void* const* d_in, const int* in_sizes, int n_in,
                              void* d_out, int out_size, void* d_ws, size_t ws_size,
                              hipStream_t stream) {
    const float* x  = (const float*)d_in[0];
    const int*   ei = (const int*)  d_in[1];
    const float* W1 = (const float*)d_in[2];
    const float* b1 = (const float*)d_in[3];
    const float* W2 = (const float*)d_in[4];
    const float* b2 = (const float*)d_in[5];
    float* out = (float*)d_out;

    const int N = in_sizes[0] / IN_F;     // 50000
    const int E = in_sizes[1] / 2;        // 800000
    const int* src = ei;
    const int* dst = ei + E;

    // workspace carve-out (~66 MB): deg, dinv, Bt1, Bt2, h1, agg1, h2, agg2
    char* ws = (char*)d_ws;
    size_t off = 0;
    auto carve = [&](size_t bytes) -> void* {
        void* p = ws + off;
        off += (bytes + 255) & ~(size_t)255;
        return p;
    };
    int*   deg  = (int*)  carve((size_t)N * sizeof(int));
    float* dinv = (float*)carve((size_t)N * sizeof(float));
    float* Bt1  = (float*)carve((size_t)HID_PAD * IN_F  * sizeof(float)); // 112x128
    float* Bt2  = (float*)carve((size_t)OUT_F   * HID_F * sizeof(float)); // 64x100
    float* h1   = (float*)carve((size_t)N * HID_F * sizeof(float));
    float* agg1 = (float*)carve((size_t)N * HID_F * sizeof(float));
    float* h2   = (float*)carve((size_t)N * OUT_F * sizeof(float));
    float* agg2 = (float*)carve((size_t)N * OUT_F * sizeof(float));
    (void)ws_size; (void)n_in; (void)out_size;

    const int rowTiles = (N + 15) / 16;   // 3125

    // degrees + rsqrt normalization
    k_deg_init   <<<(N + 255) / 256, 256, 0, stream>>>(deg, N);
    k_deg_scatter<<<(E + 255) / 256, 256, 0, stream>>>(dst, E, deg);
    k_dinv       <<<(N + 255) / 256, 256, 0, stream>>>(deg, dinv, N);

    // weight prep (tiny): transpose + pad
    k_transpose_pad<<<(HID_PAD * IN_F + 255) / 256, 256, 0, stream>>>(W1, Bt1, IN_F, HID_F, HID_PAD);
    k_transpose_pad<<<(OUT_F * HID_F + 255) / 256, 256, 0, stream>>>(W2, Bt2, HID_F, OUT_F, OUT_F);

    // layer 1: h1 = x @ W1 (WMMA f32), normalized aggregation, bias+ReLU
    k_gemm_wmma<7, IN_F, HID_F><<<rowTiles, 32, 0, stream>>>(x, Bt1, h1, N);
    {
        long long total = (long long)N * HID_F;
        int grid = (int)((total + 255) / 256);
        k_selfloop<<<grid, 256, 0, stream>>>(h1, dinv, agg1, N, HID_F);
    }
    k_edge_scatter<<<E, 128, 0, stream>>>(src, dst, h1, dinv, agg1, HID_F);
    {
        long long total = (long long)N * HID_F;
        int grid = (int)((total + 255) / 256);
        k_bias_relu<<<grid, 256, 0, stream>>>(agg1, b1, N, HID_F);
    }

    // layer 2: h2 = relu(agg1) @ W2 (WMMA f32), aggregation, bias + L2 normalize
    k_gemm_wmma<4, HID_F, OUT_F><<<rowTiles, 32, 0, stream>>>(agg1, Bt2, h2, N);
    {
        long long total = (long long)N * OUT_F;
        int grid = (int)((total + 255) / 256);
        k_selfloop<<<grid, 256, 0, stream>>>(h2, dinv, agg2, N, OUT_F);
    }
    k_edge_scatter<<<E, 64, 0, stream>>>(src, dst, h2, dinv, agg2, OUT_F);

    k_bias_norm<<<(N + 7) / 8, 256, 0, stream>>>(agg2, b2, out, N);
}